// SumDecoder_82368882803151
// MI455X (gfx1250) — compile-verified
//
#include <hip/hip_runtime.h>

typedef __attribute__((ext_vector_type(2))) float v2f;
typedef __attribute__((ext_vector_type(8))) float v8f;

#define B_   64
#define E_   256
#define H_   512
#define V_   50257
#define KV_  2048
#define G4H  2048           // 4*H gates
#define EH_  768            // E + H
#define VTAIL (V_ - KV_)    // 48209: scores land on columns [VTAIL, V)
#define NT   3141           // full 16-col tiles: 3141*16 = 50256
#define U1S  258            // LDS row stride (pad vs 256 to avoid bank conflicts)

static __device__ inline v8f v8f_zero() {
  v8f z;
  for (int i = 0; i < 8; ++i) z[i] = 0.f;
  return z;
}
static __device__ inline float sigm(float x) { return 1.f / (1.f + __expf(-x)); }
static __device__ inline float fast_tanh(float x) {
#if __has_builtin(__builtin_amdgcn_tanhf)
  return __builtin_amdgcn_tanhf(x);          // v_tanh_f32 (gfx1250 TRANS op)
#elif __has_builtin(__builtin_amdgcn_tanh_f32)
  return __builtin_amdgcn_tanh_f32(x);
#else
  return 1.f - 2.f / (__expf(2.f * x) + 1.f);  // single v_exp_f32 fallback
#endif
}

// ---------------- Kernel 0: gates = emb[tok] @ Wih.T + h0 @ Whh.T + bih + bhh (WMMA) ----
// M = 64 (4 mi tiles), N = 2048 (128 col tiles), K = 256 (x part) + 512 (h part).
// grid = 16 blocks x 8 waves, one 16-col tile per wave.
__global__ __launch_bounds__(256) void k_gates(
    const int* __restrict__ inp, const float* __restrict__ h0,
    const float* __restrict__ emb, const float* __restrict__ Wih,
    const float* __restrict__ bih, const float* __restrict__ Whh,
    const float* __restrict__ bhh, float* __restrict__ gates) {
  const int w = threadIdx.x >> 5;
  const int ct = blockIdx.x * 8 + w;            // 0..127, wave-uniform
  const int lane = threadIdx.x & 31, half = lane >> 4, li = lane & 15;
  const int n = ct * 16 + li;                   // gate row index (j)
  long tokm[4];
  for (int mi = 0; mi < 4; ++mi) tokm[mi] = inp[mi * 16 + li];
  v8f acc[4];
  for (int mi = 0; mi < 4; ++mi) acc[mi] = v8f_zero();
  const float* wi = Wih + (long)n * E_;
  const float* wh = Whh + (long)n * H_;
  // x-part: A[m][k] = emb[tok[m]][k], B[k][n] = Wih[n][k]
  for (int k0 = 0; k0 < E_; k0 += 4) {
    const int kk = k0 + half * 2;
    v2f bf; bf.x = wi[kk]; bf.y = wi[kk + 1];
    for (int mi = 0; mi < 4; ++mi) {
      const float* a = emb + tokm[mi] * E_ + kk;
      v2f af; af.x = a[0]; af.y = a[1];
      acc[mi] = __builtin_amdgcn_wmma_f32_16x16x4_f32(
          false, af, false, bf, (short)0, acc[mi], false, false);
    }
  }
  // h-part: A[m][k] = h0[m][k], B[k][n] = Whh[n][k]
  for (int k0 = 0; k0 < H_; k0 += 4) {
    const int kk = k0 + half * 2;
    v2f bf; bf.x = wh[kk]; bf.y = wh[kk + 1];
    for (int mi = 0; mi < 4; ++mi) {
      const float* a = h0 + (mi * 16 + li) * H_ + kk;
      v2f af; af.x = a[0]; af.y = a[1];
      acc[mi] = __builtin_amdgcn_wmma_f32_16x16x4_f32(
          false, af, false, bf, (short)0, acc[mi], false, false);
    }
  }
  const float bn = bih[n] + bhh[n];
  for (int mi = 0; mi < 4; ++mi)
    for (int r = 0; r < 8; ++r)
      gates[(long)(mi * 16 + half * 8 + r) * G4H + n] = acc[mi][r] + bn;
}

// ---------------- Kernel 0b: LSTM activations -> h_t, c_t ----------------
__global__ __launch_bounds__(256) void k_lstm_act(
    const float* __restrict__ gates, const float* __restrict__ c0,
    float* __restrict__ outH, float* __restrict__ outC) {
  const int idx = blockIdx.x * 256 + threadIdx.x;  // 0..32767
  const int b = idx >> 9, hh = idx & 511;
  const float* g = gates + (long)b * G4H;
  const float ig = sigm(g[hh]);
  const float fg = sigm(g[H_ + hh]);
  const float gg = fast_tanh(g[2 * H_ + hh]);
  const float og = sigm(g[3 * H_ + hh]);
  const float c = fg * c0[b * H_ + hh] + ig * gg;
  outH[b * H_ + hh] = og * fast_tanh(c);
  outC[b * H_ + hh] = c;
}

// ---------------- Kernel 1a: hproj = h_t @ W1[:, :H].T + b1 (WMMA) ----------------
// M=64 (4 mi), N=256 (16 col tiles), K=512. grid = 2 blocks x 8 waves.
__global__ __launch_bounds__(256) void k_hproj(
    const float* __restrict__ hT, const float* __restrict__ W1, const float* __restrict__ b1,
    float* __restrict__ hproj) {
  const int w = threadIdx.x >> 5;
  const int ct = blockIdx.x * 8 + w;            // 0..15
  const int lane = threadIdx.x & 31, half = lane >> 4, li = lane & 15;
  const int n = ct * 16 + li;                   // output feature e
  v8f acc[4];
  for (int mi = 0; mi < 4; ++mi) acc[mi] = v8f_zero();
  const float* wp = W1 + (long)n * EH_;         // h part: columns [0, H)
  for (int k0 = 0; k0 < H_; k0 += 4) {
    const int kk = k0 + half * 2;
    v2f bf; bf.x = wp[kk]; bf.y = wp[kk + 1];
    for (int mi = 0; mi < 4; ++mi) {
      const float* a = hT + (mi * 16 + li) * H_ + kk;
      v2f af; af.x = a[0]; af.y = a[1];
      acc[mi] = __builtin_amdgcn_wmma_f32_16x16x4_f32(
          false, af, false, bf, (short)0, acc[mi], false, false);
    }
  }
  const float bn = b1[n];
  for (int mi = 0; mi < 4; ++mi)
    for (int r = 0; r < 8; ++r)
      hproj[(mi * 16 + half * 8 + r) * E_ + n] = acc[mi][r] + bn;
}

// ---------------- Kernel 1b: kproj = k @ W1[:, H:].T (WMMA) ----------------
// M=2048 (128 mi), N=256 (16 ni), K=256. One 16x16 tile per wave; 256 blocks.
__global__ __launch_bounds__(256) void k_kproj(
    const float* __restrict__ K, const float* __restrict__ W1, float* __restrict__ kproj) {
  const int w = threadIdx.x >> 5;
  const int tile = blockIdx.x * 8 + w;          // 0..2047, wave-uniform
  const int mi = tile >> 4, ni = tile & 15;
  const int lane = threadIdx.x & 31, half = lane >> 4, li = lane & 15;
  const int n = ni * 16 + li;
  const float* wp = W1 + (long)n * EH_ + H_;    // k part: columns [H, H+E)
  const float* ap = K + (long)(mi * 16 + li) * E_;
  v8f acc = v8f_zero();
  for (int k0 = 0; k0 < E_; k0 += 4) {
    const int kk = k0 + half * 2;
    v2f bf; bf.x = wp[kk]; bf.y = wp[kk + 1];
    v2f af; af.x = ap[kk]; af.y = ap[kk + 1];
    acc = __builtin_amdgcn_wmma_f32_16x16x4_f32(
        false, af, false, bf, (short)0, acc, false, false);
  }
  for (int r = 0; r < 8; ++r)
    kproj[(long)(mi * 16 + half * 8 + r) * E_ + n] = acc[r];
}

// ---------------- Kernel 2: u1=tanh(hproj+kproj); u2=tanh(u1@W2.T+b2) via WMMA;
//                  scores = u2 @ w3 + b3.  One block = (b, 32 kv rows). ----------------
__global__ __launch_bounds__(256) void k_attn(
    const float* __restrict__ hproj, const float* __restrict__ kproj,
    const float* __restrict__ W2, const float* __restrict__ b2,
    const float* __restrict__ W3, const float* __restrict__ b3,
    float* __restrict__ scores) {
  __shared__ float u1[32 * U1S];
  __shared__ float sc[32];
  const int kv0 = blockIdx.x * 32;
  const int b = blockIdx.y;
  const int t = threadIdx.x;
  if (t < 32) sc[t] = 0.f;
  for (int idx = t; idx < 32 * 256; idx += 256) {
    const int r = idx >> 8, e = idx & 255;
    u1[r * U1S + e] = fast_tanh(hproj[b * E_ + e] + kproj[(long)(kv0 + r) * E_ + e]);
  }
  __syncthreads();

  const int w = t >> 5, lane = t & 31, half = lane >> 4, li = lane & 15;
  v8f acc[2][2];
  for (int mi = 0; mi < 2; ++mi)
    for (int ni = 0; ni < 2; ++ni) acc[mi][ni] = v8f_zero();
  const int n0 = w * 32;                         // wave owns cols [n0, n0+32)
  for (int k0 = 0; k0 < E_; k0 += 4) {
    const int kk = k0 + half * 2;
    v2f bf[2], af[2];
    for (int ni = 0; ni < 2; ++ni) {
      const float* p = W2 + (long)(n0 + ni * 16 + li) * E_ + kk;  // B[j][n] = W2[n][j]
      bf[ni].x = p[0]; bf[ni].y = p[1];
    }
    for (int mi = 0; mi < 2; ++mi) {
      const int m = mi * 16 + li;
      af[mi].x = u1[m * U1S + kk];
      af[mi].y = u1[m * U1S + kk + 1];
    }
    for (int mi = 0; mi < 2; ++mi)
      for (int ni = 0; ni < 2; ++ni)
        acc[mi][ni] = __builtin_amdgcn_wmma_f32_16x16x4_f32(
            false, af[mi], false, bf[ni], (short)0, acc[mi][ni], false, false);
  }
  for (int mi = 0; mi < 2; ++mi)
    for (int r = 0; r < 8; ++r) {
      float v = 0.f;
      for (int ni = 0; ni < 2; ++ni) {
        const int n = n0 + ni * 16 + li;
        v += fast_tanh(acc[mi][ni][r] + b2[n]) * W3[n];
      }
      atomicAdd(&sc[mi * 16 + half * 8 + r], v);
    }
  __syncthreads();
  if (t < 32) scores[b * KV_ + kv0 + t] = sc[t] + b3[0];
}

// ---------------- Kernel 3: o = h_t @ W_out.T + b_out (+scores on tail cols), WMMA ----
__global__ __launch_bounds__(256) void k_outproj(
    const float* __restrict__ hT, const float* __restrict__ Wout,
    const float* __restrict__ bout, const float* __restrict__ scores,
    float* __restrict__ o) {
  const int w = threadIdx.x >> 5;
  const int ct = blockIdx.x * 8 + w;            // 16-col tile id, wave-uniform
  if (ct >= NT) return;
  const int lane = threadIdx.x & 31, half = lane >> 4, li = lane & 15;
  const int n = ct * 16 + li;
  v8f acc[4];
  for (int mi = 0; mi < 4; ++mi) acc[mi] = v8f_zero();
  const float* wp = Wout + (long)n * H_;        // B[k][n] = W_out[n][k]
  __builtin_prefetch(wp, 0, 1);
  for (int k0 = 0; k0 < H_; k0 += 4) {
    const int kk = k0 + half * 2;
    v2f bf; bf.x = wp[kk]; bf.y = wp[kk + 1];
    for (int mi = 0; mi < 4; ++mi) {
      v2f af;
      af.x = hT[(mi * 16 + li) * H_ + kk];
      af.y = hT[(mi * 16 + li) * H_ + kk + 1];
      acc[mi] = __builtin_amdgcn_wmma_f32_16x16x4_f32(
          false, af, false, bf, (short)0, acc[mi], false, false);
    }
  }
  const float bn = bout[n];
  for (int mi = 0; mi < 4; ++mi)
    for (int r = 0; r < 8; ++r) {
      const int row = mi * 16 + half * 8 + r;
      float v = acc[mi][r] + bn;
      if (n >= VTAIL) v += scores[row * KV_ + (n - VTAIL)];
      o[(long)row * V_ + n] = v;
    }
}

// ---------------- Kernel 3b: last column (V-1), scalar ----------------
__global__ void k_outproj_tail(const float* __restrict__ hT, const float* __restrict__ Wout,
                               const float* __restrict__ bout, const float* __restrict__ scores,
                               float* __restrict__ o) {
  const int b = threadIdx.x;  // 64 threads
  const float* wp = Wout + (long)(V_ - 1) * H_;
  float acc = bout[V_ - 1];
  for (int kk = 0; kk < H_; ++kk) acc += hT[b * H_ + kk] * wp[kk];
  acc += scores[b * KV_ + (KV_ - 1)];
  o[(long)b * V_ + (V_ - 1)] = acc;
}

// ---------------- Kernel 4: in-place row-wise log_softmax ----------------
__global__ __launch_bounds__(256) void k_logsoftmax(float* __restrict__ o) {
  __shared__ float red[256];
  const int b = blockIdx.x, t = threadIdx.x;
  float* row = o + (long)b * V_;
  float m = -1e30f;
  for (int i = t; i < V_; i += 256) m = fmaxf(m, row[i]);
  red[t] = m;
  __syncthreads();
  for (int s = 128; s > 0; s >>= 1) {
    if (t < s) red[t] = fmaxf(red[t], red[t + s]);
    __syncthreads();
  }
  const float mx = red[0];
  __syncthreads();
  float s = 0.f;
  for (int i = t; i < V_; i += 256) s += __expf(row[i] - mx);
  red[t] = s;
  __syncthreads();
  for (int ss = 128; ss > 0; ss >>= 1) {
    if (t < ss) red[t] += red[t + ss];
    __syncthreads();
  }
  const float lse = mx + __logf(red[0]);
  for (int i = t; i < V_; i += 256) row[i] = row[i] - lse;
}

extern "C" void kernel_launch(void* const* d_in, const int* in_sizes, int n_in,
                              void* d_out, int out_size, void* d_ws, size_t ws_size,
                              hipStream_t stream) {
  (void)in_sizes; (void)n_in; (void)out_size; (void)ws_size;
  const int*   inp  = (const int*)  d_in[0];
  const float* h0   = (const float*)d_in[1];
  const float* c0   = (const float*)d_in[2];
  const float* kmat = (const float*)d_in[3];
  /* d_in[4] context_vector: unused by reference */
  const float* emb  = (const float*)d_in[5];
  const float* Wih  = (const float*)d_in[6];
  const float* bih  = (const float*)d_in[7];
  const float* Whh  = (const float*)d_in[8];
  const float* bhh  = (const float*)d_in[9];
  const float* W1   = (const float*)d_in[10];
  const float* b1   = (const float*)d_in[11];
  const float* W2   = (const float*)d_in[12];
  const float* b2   = (const float*)d_in[13];
  const float* W3   = (const float*)d_in[14];
  const float* b3   = (const float*)d_in[15];
  const float* Wout = (const float*)d_in[16];
  const float* bout = (const float*)d_in[17];

  float* y    = (float*)d_out;            // (B, V) logits -> log_softmax in place
  float* outH = y + (long)B_ * V_;        // (B, H)
  float* outC = outH + B_ * H_;           // (B, H)

  float* gates  = (float*)d_ws;                   // B*4H  = 131072 f32
  float* hproj  = gates + (long)B_ * G4H;         // B*E   = 16384 f32
  float* kproj  = hproj + B_ * E_;                // KV*E  = 524288 f32
  float* scores = kproj + (long)KV_ * E_;         // B*KV  = 131072 f32 (~3.2 MB total)

  k_gates<<<16, 256, 0, stream>>>(inp, h0, emb, Wih, bih, Whh, bhh, gates);
  k_lstm_act<<<(B_ * H_) / 256, 256, 0, stream>>>(gates, c0, outH, outC);
  k_hproj<<<2, 256, 0, stream>>>(outH, W1, b1, hproj);
  k_kproj<<<256, 256, 0, stream>>>(kmat, W1, kproj);
  k_attn<<<dim3(KV_ / 32, B_), 256, 0, stream>>>(hproj, kproj, W2, b2, W3, b3, scores);
  k_outproj<<<(NT + 7) / 8, 256, 0, stream>>>(outH, Wout, bout, scores, y);
  k_outproj_tail<<<1, 64, 0, stream>>>(outH, Wout, bout, scores, y);
  k_logsoftmax<<<B_, 256, 0, stream>>>(y);
}